// CasualAttention_30940944400931
// MI455X (gfx1250) — compile-verified
//
#include <hip/hip_runtime.h>
#include <math.h>
#include <stdint.h>

typedef __attribute__((ext_vector_type(2))) float    v2f;
typedef __attribute__((ext_vector_type(8))) float    v8f;
typedef __attribute__((ext_vector_type(4))) unsigned v4u;
typedef __attribute__((ext_vector_type(8))) int      v8i;
typedef __attribute__((ext_vector_type(4))) int      v4i;

#define TSEQ 2048
#define CEMB 1024
#define HDIM 16
#define NHEAD 64
#define LDQKV 3072

// ---------------------------------------------------------------------------
// Generic fp32 GEMM:  C[M,N] = A[M,K] @ B[K,N] + bias[N]
// One wave per 16x64 output strip; V_WMMA_F32_16X16X4_F32 along K.
// A-frag (16x4 f32): lane L holds row M=L%16, K = (L/16)*2 + {0,1}  (b64 load)
// B-frag (4x16 f32): lane L holds col N=L%16, K = (L/16)*2 + {0,1}  (2x b32)
// C/D   (16x16 f32): VGPR r, lane L -> row r + 8*(L/16), col L%16
// ---------------------------------------------------------------------------
__global__ __launch_bounds__(32) void gemm_wmma_f32(
    const float* __restrict__ A, const float* __restrict__ B,
    const float* __restrict__ bias, float* __restrict__ C,
    int M, int N, int K)
{
  const int lane = threadIdx.x & 31;
  const int kh   = lane >> 4;   // half-wave select
  const int l16  = lane & 15;
  const int n0   = blockIdx.x * 64;
  const int m0   = blockIdx.y * 16;

  v8f acc[4];
#pragma unroll
  for (int t = 0; t < 4; ++t)
#pragma unroll
    for (int r = 0; r < 8; ++r) acc[t][r] = 0.0f;

  const float* arow = A + (size_t)(m0 + l16) * K + kh * 2;
  for (int k = 0; k < K; k += 4) {
    v2f a = *(const v2f*)(arow + k);
    const float* brow = B + (size_t)(k + kh * 2) * N + n0 + l16;
    __builtin_prefetch(arow + k + 64, 0, 0);             // global_prefetch_b8
#pragma unroll
    for (int t = 0; t < 4; ++t) {
      v2f b;
      b.x = brow[t * 16];
      b.y = brow[(size_t)N + t * 16];
      acc[t] = __builtin_amdgcn_wmma_f32_16x16x4_f32(
          false, a, false, b, (short)0, acc[t], false, false);
    }
  }

#pragma unroll
  for (int t = 0; t < 4; ++t) {
#pragma unroll
    for (int r = 0; r < 8; ++r) {
      const int row = m0 + r + 8 * kh;
      const int col = n0 + t * 16 + l16;
      C[(size_t)row * N + col] = acc[t][r] + bias[col];
    }
  }
}

// ---------------------------------------------------------------------------
// TDM: DMA one 16x16 f32 tile (row stride LDQKV floats) from global into LDS.
// D# built per CDNA5 ISA 8.3/8.4: group0 = {count=1 | lds_addr | global_addr |
// type=2}, group1 = {data_size=4B, tensor_dim0=16, tensor_dim1=2048,
// tile_dim0=16, tile_dim1=16, tensor_dim0_stride=3072}. Groups 2/3 zero
// (<=2D tensor). Tracked on TENSORcnt.
// ---------------------------------------------------------------------------
__device__ __forceinline__ void tdm_load_tile16x16(unsigned lds_byte_off,
                                                   const void* gaddr)
{
  const unsigned long long ga = (unsigned long long)(uintptr_t)gaddr;
  v4u g0;
  g0.x = 1u;                                     // count=1, user descriptor
  g0.y = lds_byte_off;                           // lds_addr [63:32]
  g0.z = (unsigned)ga;                           // global_addr [95:64]
  g0.w = (unsigned)((ga >> 32) & 0x01FFFFFFu)    // global_addr [120:96]
       | (2u << 30);                             // type=2 ("image")
  v8i g1;
  g1[0] = 0x00020000;                            // data_size=2 -> 4 bytes
  g1[1] = (int)(16u << 16);                      // tensor_dim0 = 16 (lo16)
  g1[2] = (int)(2048u << 16);                    // tensor_dim1 = 2048 (lo16)
  g1[3] = (int)(16u << 16);                      // tile_dim0 = 16
  g1[4] = 16;                                    // tile_dim1 = 16, tile_dim2 = 0
  g1[5] = LDQKV;                                 // tensor_dim0_stride = 3072
  g1[6] = 0;
  g1[7] = 0;
  const v4i z4 = {0, 0, 0, 0};
  const v8i z8 = {0, 0, 0, 0, 0, 0, 0, 0};
  __builtin_amdgcn_tensor_load_to_lds(g0, g1, z4, z4, z8, 0);
}

// ---------------------------------------------------------------------------
// Flash-style causal attention, head_dim = 16, one wave per (16-query tile, head).
// Q/K/V are 16-column slices of the qkv buffer (row stride 3072):
//   Q = qkv[:, h*16 : h*16+16], K = +1024, V = +2048
// K/V tiles are staged into LDS by the Tensor Data Mover (double-buffered,
// s_wait_tensorcnt), online softmax, P bounced through LDS C->A layout.
// ---------------------------------------------------------------------------
__global__ __launch_bounds__(32) void flash_attn_h16(
    const float* __restrict__ qkv, float* __restrict__ out)
{
  __shared__ float kbuf[2][16 * 16];
  __shared__ float vbuf[2][16 * 16];
  __shared__ float pls[16 * 16];     // one wave per block -> private

  const int lane  = threadIdx.x & 31;
  const int kh    = lane >> 4;
  const int l16   = lane & 15;
  const int qtile = blockIdx.x;      // 0..127
  const int h     = blockIdx.y;      // 0..63
  const int qbase = qtile * 16;

  const float* Q  = qkv + (size_t)h * HDIM;
  const float* Kp = Q + CEMB;
  const float* V  = Q + 2 * CEMB;

  // LDS byte offsets for the TDM descriptors (addr[31:0] of the flat address)
  const unsigned kofs[2] = { (unsigned)(uintptr_t)&kbuf[0][0],
                             (unsigned)(uintptr_t)&kbuf[1][0] };
  const unsigned vofs[2] = { (unsigned)(uintptr_t)&vbuf[0][0],
                             (unsigned)(uintptr_t)&vbuf[1][0] };

  // Q A-fragments for all 4 K-steps of d=16 (reused over every key tile)
  v2f aq[4];
#pragma unroll
  for (int kk = 0; kk < 4; ++kk)
    aq[kk] = *(const v2f*)(Q + (size_t)(qbase + l16) * LDQKV + kk * 4 + kh * 2);

  v8f acc;
  float rowmax[8], rowsum[8];
#pragma unroll
  for (int r = 0; r < 8; ++r) { acc[r] = 0.f; rowmax[r] = -3.0e38f; rowsum[r] = 0.f; }

  // prime the pipeline: DMA tile 0
  tdm_load_tile16x16(kofs[0], Kp);
  tdm_load_tile16x16(vofs[0], V);

  for (int j = 0; j <= qtile; ++j) {
    const int cur = j & 1;

    // issue DMA for tile j+1, then wait for tile j (<=2 ops still in flight)
    if (j < qtile) {
      const int nb = (j + 1) & 1;
      const size_t nrow = (size_t)(j + 1) * 16 * LDQKV;
      tdm_load_tile16x16(kofs[nb], Kp + nrow);
      tdm_load_tile16x16(vofs[nb], V + nrow);
      __builtin_amdgcn_s_wait_tensorcnt(2);
    } else {
      __builtin_amdgcn_s_wait_tensorcnt(0);
    }
    asm volatile("" ::: "memory");   // keep LDS reads below the wait

    // S = Q @ K^T : B[d][key] = K[key][d], key = l16 row of the LDS tile
    v2f bk[4];
#pragma unroll
    for (int kk = 0; kk < 4; ++kk)
      bk[kk] = *(const v2f*)(&kbuf[cur][l16 * 16 + kk * 4 + kh * 2]);

    v8f S;
#pragma unroll
    for (int r = 0; r < 8; ++r) S[r] = 0.f;
#pragma unroll
    for (int kk = 0; kk < 4; ++kk)
      S = __builtin_amdgcn_wmma_f32_16x16x4_f32(
          false, aq[kk], false, bk[kk], (short)0, S, false, false);

    const bool diag = (j == qtile);
    float pv[8];
#pragma unroll
    for (int r = 0; r < 8; ++r) {
      float s = S[r] * 0.25f;                       // 1/sqrt(16)
      if (diag && (l16 > r + 8 * kh)) s = -3.0e38f; // causal mask in diagonal tile
      // row max across the 16 lanes holding this row (xor butterfly stays in half)
      float m = s;
      m = fmaxf(m, __shfl_xor(m, 1, 32));
      m = fmaxf(m, __shfl_xor(m, 2, 32));
      m = fmaxf(m, __shfl_xor(m, 4, 32));
      m = fmaxf(m, __shfl_xor(m, 8, 32));
      const float mnew = fmaxf(rowmax[r], m);
      const float corr = __expf(rowmax[r] - mnew);
      const float p    = __expf(s - mnew);
      float ps = p;
      ps += __shfl_xor(ps, 1, 32);
      ps += __shfl_xor(ps, 2, 32);
      ps += __shfl_xor(ps, 4, 32);
      ps += __shfl_xor(ps, 8, 32);
      rowsum[r] = rowsum[r] * corr + ps;
      rowmax[r] = mnew;
      acc[r]   *= corr;
      pv[r]     = p;
    }

    // C-layout -> A-layout transpose of P through LDS (in-wave, DS ops in-order)
#pragma unroll
    for (int r = 0; r < 8; ++r)
      pls[(r + 8 * kh) * 16 + l16] = pv[r];
    asm volatile("s_wait_dscnt 0" ::: "memory");

    // acc += P @ V   (V B-frag from the LDS tile: B[key][d])
#pragma unroll
    for (int kk = 0; kk < 4; ++kk) {
      v2f ap = *(const v2f*)(&pls[l16 * 16 + kk * 4 + kh * 2]);
      v2f bv;
      bv.x = vbuf[cur][(kk * 4 + kh * 2) * 16 + l16];
      bv.y = vbuf[cur][(kk * 4 + kh * 2 + 1) * 16 + l16];
      acc = __builtin_amdgcn_wmma_f32_16x16x4_f32(
          false, ap, false, bv, (short)0, acc, false, false);
    }
  }

  // normalize and write attn output in [T, C] layout (head h -> cols h*16..h*16+15)
#pragma unroll
  for (int r = 0; r < 8; ++r) {
    const float inv = 1.0f / rowsum[r];
    out[(size_t)(qbase + r + 8 * kh) * CEMB + h * HDIM + l16] = acc[r] * inv;
  }
}

// ---------------------------------------------------------------------------
extern "C" void kernel_launch(void* const* d_in, const int* in_sizes, int n_in,
                              void* d_out, int out_size, void* d_ws, size_t ws_size,
                              hipStream_t stream) {
  (void)in_sizes; (void)n_in; (void)out_size; (void)ws_size;

  const float* x     = (const float*)d_in[0];
  // d_in[1] = attn_mask (ignored: is_causal wins in the reference)
  const float* W_qkv = (const float*)d_in[2];
  const float* b_qkv = (const float*)d_in[3];
  const float* W_out = (const float*)d_in[4];
  const float* b_out = (const float*)d_in[5];
  float* outp = (float*)d_out;

  float* qkv     = (float*)d_ws;                     // [2048, 3072] = 24 MiB
  float* attnbuf = qkv + (size_t)TSEQ * LDQKV;       // [2048, 1024] =  8 MiB

  dim3 blk(32, 1, 1);

  // 1) qkv = x @ W_qkv + b_qkv
  gemm_wmma_f32<<<dim3(LDQKV / 64, TSEQ / 16), blk, 0, stream>>>(
      x, W_qkv, b_qkv, qkv, TSEQ, LDQKV, CEMB);

  // 2) per-head causal attention (64 heads, d=16)
  flash_attn_h16<<<dim3(TSEQ / 16, NHEAD), blk, 0, stream>>>(qkv, attnbuf);

  // 3) out = attn @ W_out + b_out
  gemm_wmma_f32<<<dim3(CEMB / 64, TSEQ / 16), blk, 0, stream>>>(
      attnbuf, W_out, b_out, outp, TSEQ, CEMB, CEMB);
}